// MeanEncoder_23081154249147
// MI455X (gfx1250) — compile-verified
//
#include <hip/hip_runtime.h>
#include <stdint.h>
#include <math.h>

// Problem constants (match the reference).
#define KVOCAB  50000
#define KEMB    128
#define KRADIUS 3
#define KREGION 7          // 2*KRADIUS + 1
#define KB      16
#define KL      2048
#define KPOSPB  8          // positions processed per (single-wave) block
#define KTOK    (KREGION + KPOSPB - 1)   // distinct window tokens per block = 14

static_assert(KL % KPOSPB == 0, "block must not straddle a batch boundary");

__device__ __forceinline__ float fast_tanh(float x) {
#if __has_builtin(__builtin_amdgcn_tanhf)
    return __builtin_amdgcn_tanhf(x);          // v_tanh_f32 (TRANS)
#elif __has_builtin(__builtin_amdgcn_tanh_f32)
    return __builtin_amdgcn_tanh_f32(x);       // v_tanh_f32 (TRANS)
#else
    return tanhf(x);                           // ocml fallback
#endif
}

// One wave32 per 8 positions; lane owns 4 embedding columns (float4).
// Rows are staged global->LDS with CDNA5 async-to-LDS ops (ASYNCcnt),
// double-buffered across positions so position p+1's 7 rows are in flight
// while position p is reduced out of LDS.
__global__ __launch_bounds__(32) void
MeanEncoder_23081154249147_kernel(const int* __restrict__ seq,
                                  const float* __restrict__ W,
                                  float* __restrict__ out) {
    // Double buffer: 2 * 7 rows * 128 f32 = 7168 B. Single shared array ->
    // placed at LDS offset 0, so arithmetic byte offsets match the asm writes.
    __shared__ __align__(16) float lds[2 * KREGION * KEMB];
    float* lds_escape = lds;   // escapes into asm so "memory" clobbers cover it

    const int lane = threadIdx.x;                 // 0..31
    const int pos0 = blockIdx.x * KPOSPB;         // first flat position (b*L + l)
    const int b    = pos0 / KL;                   // whole block in one batch
    const int l0   = pos0 - b * KL;
    const int* srow = seq + b * KL;

    // Load the 14 distinct window tokens once, branchlessly (uniform -> SALU).
    // Out-of-range positions behave like jnp.pad zeros: token = 0.
    int tok[KTOK];
    #pragma unroll
    for (int j = 0; j < KTOK; ++j) {
        const int q  = l0 + j - KRADIUS;
        const int qc = (q < 0) ? 0 : ((q > KL - 1) ? KL - 1 : q);  // clamp addr
        const int v  = srow[qc];                                   // safe load
        tok[j] = (q == qc) ? v : 0;                                // csel OOB->0
    }

    // Per-lane global base: each lane owns a 16 B slice of every row.
    const uint64_t vbase = (uint64_t)(uintptr_t)(W) + (uint64_t)(lane * 16);

    // Issue the 7 async row copies for position (pos0+p) into buffer `buf`.
    auto issue = [&](int p, int buf) {
        #pragma unroll
        for (int i = 0; i < KREGION; ++i) {
            // Row byte offset: (tok*7 + i) * 128 floats * 4 B = (tok*7+i)*512.
            const uint32_t roff = (uint32_t)(tok[p + i] * KREGION + i) * 512u;
            const uint64_t ga   = vbase + roff;
            const uint32_t loff =
                (uint32_t)(((buf * KREGION + i) * KEMB + lane * 4) * 4);
            // CDNA5: async global->LDS copy, tracked with ASYNCcnt.
            // %2 is unreferenced: it only makes `lds` escape into the asm.
            asm volatile("global_load_async_to_lds_b128 %0, %1, off"
                         :: "v"(loff), "v"(ga), "s"(lds_escape)
                         : "memory");
        }
    };

    issue(0, 0);  // prime the pipeline

    #pragma unroll
    for (int p = 0; p < KPOSPB; ++p) {
        const int buf = p & 1;
        if (p + 1 < KPOSPB) {
            issue(p + 1, buf ^ 1);
            // 14 outstanding; async loads complete in order, so <=7 means
            // this position's 7 rows have landed in LDS.
            asm volatile("s_wait_asynccnt 0x7" ::: "memory");
        } else {
            asm volatile("s_wait_asynccnt 0x0" ::: "memory");
        }

        // Reduce the 7 rows for this position (ds_load_b128 per row).
        float a0 = 0.f, a1 = 0.f, a2 = 0.f, a3 = 0.f;
        #pragma unroll
        for (int i = 0; i < KREGION; ++i) {
            const float4 v =
                *(const float4*)&lds[(buf * KREGION + i) * KEMB + lane * 4];
            a0 += v.x; a1 += v.y; a2 += v.z; a3 += v.w;
        }

        // Center-token mask: tok[p+RADIUS] is exactly seq[pos0+p] (in range).
        const float m = (tok[p + KRADIUS] != 0) ? 1.0f : 0.0f;
        float4 r;
        r.x = fast_tanh(a0) * m;
        r.y = fast_tanh(a1) * m;
        r.z = fast_tanh(a2) * m;
        r.w = fast_tanh(a3) * m;
        *(float4*)(out + (long long)(pos0 + p) * KEMB + lane * 4) = r;
    }
}

extern "C" void kernel_launch(void* const* d_in, const int* in_sizes, int n_in,
                              void* d_out, int out_size, void* d_ws, size_t ws_size,
                              hipStream_t stream) {
    (void)in_sizes; (void)n_in; (void)d_ws; (void)ws_size; (void)out_size;
    const int*   seq = (const int*)d_in[0];    // [B, L, 1] int32
    const float* W   = (const float*)d_in[1];  // [VOCAB*REGION, EMB] f32
    float*       out = (float*)d_out;          // [B, L, 1, EMB] f32

    const int total_pos = KB * KL;             // 32768
    dim3 grid(total_pos / KPOSPB);             // 4096 single-wave blocks
    dim3 block(32);
    MeanEncoder_23081154249147_kernel<<<grid, block, 0, stream>>>(seq, W, out);
}